// PotentialLoss_82386062672209
// MI455X (gfx1250) — compile-verified
//
#include <hip/hip_runtime.h>
#include <hip/hip_bf16.h>
#include <math.h>

#define Q_MIN 0.01f
#define REP_SCALE 10.0f
#define PNUM 512
#define DDIM 8
#define TM 16
#define WAVES_PER_BLOCK 8
#define BLOCK 256

typedef float v2f __attribute__((ext_vector_type(2)));
typedef float v8f __attribute__((ext_vector_type(8)));

// raw v_sqrt_f32 (TRANS32), no IEEE fixup sequence; inputs are clamped >= 0
__device__ __forceinline__ float fast_sqrtf(float v) {
#if __has_builtin(__builtin_amdgcn_sqrtf)
  return __builtin_amdgcn_sqrtf(v);
#else
  return sqrtf(v);
#endif
}

// ---------------- phase 1: per-particle segment stats ----------------

__global__ void pl_init(unsigned* qmax_bits, int* counts, int* alpha_idx) {
  int p = blockIdx.x * blockDim.x + threadIdx.x;
  if (p < PNUM) {
    qmax_bits[p] = 0u;          // q >= Q_MIN > 0, so uint-compare == float-compare
    counts[p]    = 0;
    alpha_idx[p] = 0x7fffffff;
  }
}

__global__ void pl_hits_pass1(const float* __restrict__ beta,
                              const float* __restrict__ x,
                              const int*   __restrict__ pid,
                              float* __restrict__ q, float* __restrict__ xx,
                              unsigned* __restrict__ qmax_bits,
                              int* __restrict__ counts, int n) {
  int i = blockIdx.x * blockDim.x + threadIdx.x;
  if (i >= n) return;
  float a  = atanhf(beta[i]);
  float qi = a * a + Q_MIN;
  q[i] = qi;
  float s = 0.f;
#pragma unroll
  for (int k = 0; k < DDIM; ++k) { float v = x[i * DDIM + k]; s += v * v; }
  xx[i] = s;
  int p = pid[i];
  atomicMax(&qmax_bits[p], __float_as_uint(qi));
  atomicAdd(&counts[p], 1);
}

__global__ void pl_hits_pass2(const int* __restrict__ pid,
                              const float* __restrict__ q,
                              const unsigned* __restrict__ qmax_bits,
                              int* __restrict__ alpha_idx, int n) {
  int i = blockIdx.x * blockDim.x + threadIdx.x;
  if (i >= n) return;
  int p = pid[i];
  if (__float_as_uint(q[i]) == qmax_bits[p]) atomicMin(&alpha_idx[p], i);
}

__global__ void pl_build_alpha(const float* __restrict__ x,
                               const unsigned* __restrict__ qmax_bits,
                               const int* __restrict__ counts,
                               const int* __restrict__ alpha_idx,
                               float* __restrict__ qa, float* __restrict__ aa,
                               float* __restrict__ xa, int n) {
  int p = blockIdx.x * blockDim.x + threadIdx.x;
  if (p >= PNUM) return;
  bool valid = (counts[p] > 0) && (p != 0);
  if (valid) {
    int idx = alpha_idx[p];
    if (idx > n - 1) idx = n - 1;
    float s = 0.f;
#pragma unroll
    for (int k = 0; k < DDIM; ++k) {
      float v = x[idx * DDIM + k];
      xa[p * DDIM + k] = v;
      s += v * v;
    }
    aa[p] = s;
    qa[p] = __uint_as_float(qmax_bits[p]);
  } else {
    // q_alpha == 0 makes both attractive and repulsive terms vanish exactly
#pragma unroll
    for (int k = 0; k < DDIM; ++k) xa[p * DDIM + k] = 0.f;
    aa[p] = 0.f;
    qa[p] = 0.f;
  }
}

// ---------------- phase 2: fused WMMA pairwise kernel ----------------
// Tile: 16 hits (M) x 16 particles (N), K = D = 8 via two V_WMMA_F32_16X16X4_F32.
// A layout (32-bit A 16x4): lane<16 -> M=lane, VGPR0=K0,VGPR1=K1; lane>=16 -> K2/K3.
// B layout mirrors with N=lane%16. C/D: lane -> N=lane%16, VGPR r -> M=r+8*(lane>=16).

__global__ void __launch_bounds__(BLOCK)
pl_main(const float* __restrict__ x, const int* __restrict__ pid,
        const float* __restrict__ q, const float* __restrict__ xx,
        const float* __restrict__ qa_g, const float* __restrict__ aa_g,
        const float* __restrict__ xa_g, float* __restrict__ partials,
        int n, int ntiles) {
  __shared__ float s_xa[PNUM * DDIM];
  __shared__ float s_qa[PNUM];
  __shared__ float s_aa[PNUM];
  __shared__ float s_wsum[WAVES_PER_BLOCK];

  for (int idx = threadIdx.x; idx < PNUM * DDIM; idx += BLOCK) s_xa[idx] = xa_g[idx];
  for (int idx = threadIdx.x; idx < PNUM; idx += BLOCK) {
    s_qa[idx] = qa_g[idx];
    s_aa[idx] = aa_g[idx];
  }
  __syncthreads();

  const int lane = threadIdx.x & 31;
  const int wave = threadIdx.x >> 5;
  const int half = lane >> 4;     // 0: K pair (0,1)/(4,5), 1: K pair (2,3)/(6,7)
  const int col  = lane & 15;     // M for A, N for B/C
  const int kb   = half * 2;

  float acc = 0.f;
  const int tile = blockIdx.x * WAVES_PER_BLOCK + wave;  // wave-uniform branch below
  if (tile < ntiles) {
    const int i0 = tile * TM;

    // A-matrix fragments (fp32, 2 VGPRs each)
    int ia = i0 + col;
    if (ia > n - 1) ia = n - 1;
    v2f a0, a1;
    a0.x = x[ia * DDIM + kb];     a0.y = x[ia * DDIM + kb + 1];
    a1.x = x[ia * DDIM + kb + 4]; a1.y = x[ia * DDIM + kb + 5];

    // per-row (M) epilogue data: this lane owns M = half*8 + r
    const int mbase = i0 + half * 8;
    float qv[8], xxv[8];
    int pidv[8];
#pragma unroll
    for (int r = 0; r < 8; ++r) {
      int m = mbase + r;
      bool ok = m < n;
      int mc = ok ? m : n - 1;
      qv[r]   = ok ? q[mc] : 0.f;   // q==0 zeroes any tail contribution
      xxv[r]  = xx[mc];
      pidv[r] = ok ? pid[mc] : -1;
    }

    for (int pt = 0; pt < PNUM / TM; ++pt) {
      const int pcol = pt * TM + col;   // global particle for this lane's column
      v2f b0, b1;
      b0.x = s_xa[pcol * DDIM + kb];     b0.y = s_xa[pcol * DDIM + kb + 1];
      b1.x = s_xa[pcol * DDIM + kb + 4]; b1.y = s_xa[pcol * DDIM + kb + 5];
      const float qa_n = s_qa[pcol];
      const float aa_n = s_aa[pcol];

      v8f d = {0.f, 0.f, 0.f, 0.f, 0.f, 0.f, 0.f, 0.f};
      d = __builtin_amdgcn_wmma_f32_16x16x4_f32(false, a0, false, b0,
                                                (short)0, d, false, false);
      d = __builtin_amdgcn_wmma_f32_16x16x4_f32(false, a1, false, b1,
                                                (short)0, d, false, false);

#pragma unroll
      for (int r = 0; r < 8; ++r) {
        // sq = ||x_m||^2 - 2*dot + ||xa_n||^2, clamped at 0
        float sq = fmaf(-2.f, d[r], xxv[r] + aa_n);
        sq = fmaxf(sq, 0.f);
        float dist  = fast_sqrtf(sq);                 // single v_sqrt_f32
        float hinge = fmaxf(1.f - dist, 0.f);
        float term  = (pidv[r] == pcol) ? sq : REP_SCALE * hinge;
        acc = fmaf(qv[r] * qa_n, term, acc);
      }
    }
  }

  // deterministic wave tree reduction (wave32)
#pragma unroll
  for (int off = 16; off > 0; off >>= 1) acc += __shfl_xor(acc, off, 32);
  if (lane == 0) s_wsum[wave] = acc;
  __syncthreads();
  if (threadIdx.x == 0) {
    float s = 0.f;
#pragma unroll
    for (int wv = 0; wv < WAVES_PER_BLOCK; ++wv) s += s_wsum[wv];
    partials[blockIdx.x] = s;
  }
}

__global__ void pl_final(const float* __restrict__ partials, int nb, float inv_n,
                         float* __restrict__ out) {
  if (blockIdx.x == 0 && threadIdx.x == 0) {
    float s = 0.f;
    for (int i = 0; i < nb; ++i) s += partials[i];  // fixed order => deterministic
    out[0] = s * inv_n;
  }
}

// ---------------- launch ----------------

extern "C" void kernel_launch(void* const* d_in, const int* in_sizes, int n_in,
                              void* d_out, int out_size, void* d_ws, size_t ws_size,
                              hipStream_t stream) {
  (void)n_in; (void)out_size; (void)ws_size;
  const float* beta = (const float*)d_in[1];
  const float* x    = (const float*)d_in[2];
  const int*   pid  = (const int*)d_in[4];
  const int n = in_sizes[0];
  float* out = (float*)d_out;

  char* base = (char*)d_ws;
  size_t off = 0;
  auto alloc = [&](size_t bytes) -> char* {
    char* p = base + off;
    off = (off + bytes + 255) & ~(size_t)255;
    return p;
  };
  float*    q      = (float*)alloc((size_t)n * 4);
  float*    xx     = (float*)alloc((size_t)n * 4);
  unsigned* qmaxb  = (unsigned*)alloc(PNUM * 4);
  int*      counts = (int*)alloc(PNUM * 4);
  int*      alphai = (int*)alloc(PNUM * 4);
  float*    qa     = (float*)alloc(PNUM * 4);
  float*    aa     = (float*)alloc(PNUM * 4);
  float*    xa     = (float*)alloc(PNUM * DDIM * 4);

  const int ntiles  = (n + TM - 1) / TM;
  const int nblocks = (ntiles + WAVES_PER_BLOCK - 1) / WAVES_PER_BLOCK;
  float* partials = (float*)alloc((size_t)nblocks * 4);

  const int hb = (n + 255) / 256;
  pl_init<<<(PNUM + 255) / 256, 256, 0, stream>>>(qmaxb, counts, alphai);
  pl_hits_pass1<<<hb, 256, 0, stream>>>(beta, x, pid, q, xx, qmaxb, counts, n);
  pl_hits_pass2<<<hb, 256, 0, stream>>>(pid, q, qmaxb, alphai, n);
  pl_build_alpha<<<(PNUM + 255) / 256, 256, 0, stream>>>(x, qmaxb, counts, alphai,
                                                         qa, aa, xa, n);
  pl_main<<<nblocks, BLOCK, 0, stream>>>(x, pid, q, xx, qa, aa, xa, partials,
                                         n, ntiles);
  pl_final<<<1, 32, 0, stream>>>(partials, nblocks, 1.0f / (float)n, out);
}